// MatrixFactorization_87557203296997
// MI455X (gfx1250) — compile-verified
//
#include <hip/hip_runtime.h>

typedef __attribute__((ext_vector_type(2))) float v2f;
typedef __attribute__((ext_vector_type(4))) float v4f;
typedef __attribute__((ext_vector_type(8))) float v8f;

#define U_DIM 8192
#define I_DIM 8192
#define K_DIM 128   // WMMA loop assumes K_DIM % 4 == 0 (reference: 128)

// ---------------------------------------------------------------------------
// 1) Zero the dense score block (256 MB) with b128 stores — bandwidth floor.
// ---------------------------------------------------------------------------
__global__ void mf_zero_f4(float* __restrict__ p, long long n) {
    long long i = ((long long)blockIdx.x * blockDim.x + threadIdx.x) * 4;
    if (i < n) {
        v4f z = {0.0f, 0.0f, 0.0f, 0.0f};
        *(v4f*)(p + i) = z;
    }
}

// ---------------------------------------------------------------------------
// 2) Row-presence flags (re-initialized every call; ws is scratch).
// ---------------------------------------------------------------------------
__global__ void mf_zero_flags(unsigned* __restrict__ f, int n) {
    int i = blockIdx.x * blockDim.x + threadIdx.x;
    if (i < n) f[i] = 0u;
}

__global__ void mf_scatter_flags(const int* __restrict__ uidx,
                                 const int* __restrict__ iidx,
                                 unsigned* __restrict__ fu,
                                 unsigned* __restrict__ fi, int n) {
    int i = blockIdx.x * blockDim.x + threadIdx.x;
    if (i < n) {
        fu[uidx[i]] = 1u;   // duplicate writes of the same value: benign
        fi[iidx[i]] = 1u;
    }
}

// dst[idx] = flags[idx >> shift] ? src[idx] : 0   (shift=7 for K=128 rows, 0 for biases)
__global__ void mf_masked_fill(const float* __restrict__ src,
                               const unsigned* __restrict__ flags,
                               float* __restrict__ dst,
                               long long n, int shift) {
    long long i = (long long)blockIdx.x * blockDim.x + threadIdx.x;
    if (i < n) dst[i] = flags[i >> shift] ? src[i] : 0.0f;
}

// ---------------------------------------------------------------------------
// 3) Sparse score scatter: one wave32 computes 16 (u,i) pairs via
//    V_WMMA_F32_16X16X4_F32 (K accumulated over 32 steps), then extracts the
//    diagonal of the 16x16 f32 D tile and scatter-stores dot+biases.
//
//    A (16x4 f32) layout: lane m (0-15) holds row M=m, K=k0..k0+1 in the two
//    A VGPRs; lanes 16-31 hold K=k0+2..k0+3 of the same rows.  B (4x16) is the
//    symmetric layout with N across lanes.  Both reduce to: lane l loads the
//    contiguous float2 at column k0 + 2*(l>>4) of row idx[base + (l&15)].
// ---------------------------------------------------------------------------
__global__ void mf_score_wmma(const int* __restrict__ uidx,
                              const int* __restrict__ iidx,
                              const float* __restrict__ Umat,
                              const float* __restrict__ Imat,
                              const float* __restrict__ ub,
                              const float* __restrict__ ib,
                              float* __restrict__ out1, int ngroups) {
    int wave = (int)((blockIdx.x * (long long)blockDim.x + threadIdx.x) >> 5);
    int lane = threadIdx.x & 31;
    if (wave >= ngroups) return;          // wave-uniform; EXEC stays all-1 below

    int base = wave * 16;
    int m    = lane & 15;
    int hk   = (lane >> 4) * 2;           // 0 for lanes 0-15, 2 for lanes 16-31

    const float* ua = Umat + (long long)uidx[base + m] * K_DIM + hk;
    const float* ia = Imat + (long long)iidx[base + m] * K_DIM + hk;

    v8f c = {0.0f, 0.0f, 0.0f, 0.0f, 0.0f, 0.0f, 0.0f, 0.0f};
#pragma unroll
    for (int s = 0; s < K_DIM / 4; ++s) {
        v2f a = *(const v2f*)(ua + 4 * s);
        v2f b = *(const v2f*)(ia + 4 * s);
        // (neg_a, A, neg_b, B, c_mod, C, reuse_a, reuse_b)
        c = __builtin_amdgcn_wmma_f32_16x16x4_f32(false, a, false, b,
                                                  (short)0, c, false, false);
    }

    // Diagonal of D: p in 0..7  -> VGPR p,   lane p
    //               p in 8..15 -> VGPR p-8, lane p+16
    // In both cases the wanted VGPR index is (lane & 7).
    int r = lane & 7;
    float d = c[0];
    d = (r == 1) ? c[1] : d;
    d = (r == 2) ? c[2] : d;
    d = (r == 3) ? c[3] : d;
    d = (r == 4) ? c[4] : d;
    d = (r == 5) ? c[5] : d;
    d = (r == 6) ? c[6] : d;
    d = (r == 7) ? c[7] : d;

    bool active = (lane < 8) || (lane >= 24);
    int p = (lane < 8) ? lane : (lane - 16);
    if (active) {
        int u = uidx[base + p];
        int i = iidx[base + p];
        out1[(long long)u * I_DIM + i] = d + ub[u] + ib[i];
    }
}

// Scalar fallback for a non-multiple-of-16 tail (not hit for N=65536).
__global__ void mf_score_tail(const int* __restrict__ uidx,
                              const int* __restrict__ iidx,
                              const float* __restrict__ Umat,
                              const float* __restrict__ Imat,
                              const float* __restrict__ ub,
                              const float* __restrict__ ib,
                              float* __restrict__ out1, int start, int n) {
    int t = start + blockIdx.x * blockDim.x + threadIdx.x;
    if (t < n) {
        int u = uidx[t], i = iidx[t];
        const float* ur = Umat + (long long)u * K_DIM;
        const float* ir = Imat + (long long)i * K_DIM;
        float s = 0.0f;
        for (int k = 0; k < K_DIM; ++k) s += ur[k] * ir[k];
        out1[(long long)u * I_DIM + i] = s + ub[u] + ib[i];
    }
}

extern "C" void kernel_launch(void* const* d_in, const int* in_sizes, int n_in,
                              void* d_out, int out_size, void* d_ws, size_t ws_size,
                              hipStream_t stream) {
    const int*   uidx = (const int*)d_in[0];
    const int*   iidx = (const int*)d_in[1];
    const float* Umat = (const float*)d_in[2];
    const float* Imat = (const float*)d_in[3];
    const float* ub   = (const float*)d_in[4];
    const float* ib   = (const float*)d_in[5];
    float* out = (float*)d_out;
    const int N = in_sizes[0];

    const long long O1  = (long long)U_DIM * I_DIM;     // dense score block
    const long long NUK = (long long)U_DIM * K_DIM;
    const long long NIK = (long long)I_DIM * K_DIM;
    float* out1 = out;
    float* out2 = out1 + O1;          // user_factor masked
    float* out3 = out2 + NUK;         // user_bias masked
    float* out4 = out3 + U_DIM;       // item_factor masked
    float* out5 = out4 + NIK;         // item_bias masked

    unsigned* flag_u = (unsigned*)d_ws;
    unsigned* flag_i = flag_u + U_DIM;

    // 1) zero dense score block (b128 stores)
    {
        long long n4 = (O1 + 3) / 4;
        int blocks = (int)((n4 + 255) / 256);
        mf_zero_f4<<<blocks, 256, 0, stream>>>(out1, O1);
    }
    // 2) flags: zero then scatter
    mf_zero_flags<<<(U_DIM + I_DIM + 255) / 256, 256, 0, stream>>>(flag_u, U_DIM + I_DIM);
    mf_scatter_flags<<<(N + 255) / 256, 256, 0, stream>>>(uidx, iidx, flag_u, flag_i, N);

    // 3) masked copies (outputs 2..5, fully overwritten each call)
    mf_masked_fill<<<(int)((NUK + 255) / 256), 256, 0, stream>>>(Umat, flag_u, out2, NUK, 7);
    mf_masked_fill<<<(U_DIM + 255) / 256, 256, 0, stream>>>(ub, flag_u, out3, U_DIM, 0);
    mf_masked_fill<<<(int)((NIK + 255) / 256), 256, 0, stream>>>(Imat, flag_i, out4, NIK, 7);
    mf_masked_fill<<<(I_DIM + 255) / 256, 256, 0, stream>>>(ib, flag_i, out5, I_DIM, 0);

    // 4) sparse WMMA score scatter: 16 pairs per wave32
    int ngroups = N / 16;
    if (ngroups > 0) {
        long long threads = (long long)ngroups * 32;
        int blocks = (int)((threads + 255) / 256);
        mf_score_wmma<<<blocks, 256, 0, stream>>>(uidx, iidx, Umat, Imat, ub, ib, out1, ngroups);
    }
    int done = ngroups * 16;
    if (done < N) {
        mf_score_tail<<<(N - done + 255) / 256, 256, 0, stream>>>(uidx, iidx, Umat, Imat, ub, ib, out1, done, N);
    }
}